// PathsLoss_54168127537190
// MI455X (gfx1250) — compile-verified
//
#include <hip/hip_runtime.h>
#include <cmath>

// Problem constants (fixed by the reference)
#define N      2048
#define D      32
#define PAIRS  5       // T-1 consecutive pairs
#define SPLITS 8       // reduction-axis split for parallelism
#define NITERS 80

typedef __attribute__((ext_vector_type(2))) float v2f;
typedef __attribute__((ext_vector_type(8))) float v8f;

// ---------------------------------------------------------------------------
// Online log-sum-exp (base-2) push: m = running max, s = running sum of 2^(v-m)
// Branch-free; exp2f lowers to native v_exp_f32 (hardware is base-2).
// ---------------------------------------------------------------------------
__device__ __forceinline__ void lse_push(float v, float& m, float& s) {
  float nm = fmaxf(m, v);
  s = fmaf(s, exp2f(m - nm), exp2f(v - nm));
  m = nm;
}

// ---------------------------------------------------------------------------
// Zero the dual potentials f, g (harness poisons workspace with 0xAA).
// ---------------------------------------------------------------------------
__global__ void init_fg(float* __restrict__ f, float* __restrict__ g) {
  int i = blockIdx.x * blockDim.x + threadIdx.x;
  if (i < PAIRS * N) { f[i] = 0.f; g[i] = 0.f; }
}

// ---------------------------------------------------------------------------
// Build C[pair][i][j] = 0.5*(|x_i|^2 + |y_j|^2) - <x_i, y_j> via fp32 WMMA.
// One wave computes one 16x16 tile with 8x V_WMMA_F32_16X16X4_F32 (K=32).
// A (16x4 f32): lane l(0..15)/l+16 holds A[l][2*hi + {0,1}] in v0/v1 (hi=lane>>4)
// B (4x16 f32): symmetric layout with N in place of M.
// D (16x16 f32): acc[r] @ lane -> row = 8*hi + r, col = lane&15.
// ---------------------------------------------------------------------------
__global__ __launch_bounds__(256) void build_cost(const float* __restrict__ x,
                                                  float* __restrict__ C) {
  const int pair = blockIdx.y;
  const float* __restrict__ X = x + (size_t)pair * N * D;
  const float* __restrict__ Y = X + (size_t)N * D;
  float* __restrict__ Cp = C + (size_t)pair * N * N;

  const int wave = threadIdx.x >> 5;
  const int lane = threadIdx.x & 31;
  const int tile = blockIdx.x * 8 + wave;          // 0..16383
  const int i0 = (tile >> 7) << 4;
  const int j0 = (tile & 127) << 4;
  const int l  = lane & 15;
  const int hi = lane >> 4;

  const float* xr = X + (size_t)(i0 + l) * D;
  const float* yr = Y + (size_t)(j0 + l) * D;

  v8f acc = {0.f, 0.f, 0.f, 0.f, 0.f, 0.f, 0.f, 0.f};
#pragma unroll
  for (int k0 = 0; k0 < D; k0 += 4) {
    v2f a, b;
    a.x = xr[k0 + 2 * hi];
    a.y = xr[k0 + 2 * hi + 1];
    b.x = yr[k0 + 2 * hi];
    b.y = yr[k0 + 2 * hi + 1];
    acc = __builtin_amdgcn_wmma_f32_16x16x4_f32(false, a, false, b,
                                                (short)0, acc, false, false);
  }

  // 0.5*|x_row|^2 and 0.5*|y_col|^2 (each lane computes its l-th row/col)
  float sx = 0.f, sy = 0.f;
#pragma unroll
  for (int k = 0; k < D; ++k) {
    float xv = xr[k]; sx = fmaf(xv, xv, sx);
    float yv = yr[k]; sy = fmaf(yv, yv, sy);
  }
  sx *= 0.5f; sy *= 0.5f;

#pragma unroll
  for (int r = 0; r < 8; ++r) {
    int row = hi * 8 + r;
    float sxr = __shfl(sx, row, 32);               // broadcast row-norm
    Cp[(size_t)(i0 + row) * N + j0 + l] = sxr + sy - acc[r];
  }
}

// ---------------------------------------------------------------------------
// g-update partial: for column j, reduce rows i in [i0, i0+256).
// v2 = (f_i - C_ij)*k2 + log_a*log2e = fk[i] - k2*C_ij   (base-2 domain)
// Thread t owns column j0+t -> fully coalesced reads of C.
// ---------------------------------------------------------------------------
__global__ __launch_bounds__(256) void g_partial_k(
    const float* __restrict__ C, const float* __restrict__ f,
    float* __restrict__ pm, float* __restrict__ ps, float k2, float c2) {
  __shared__ float fk[256];
  const int pair = blockIdx.z;
  const float* __restrict__ Cp = C + (size_t)pair * N * N;
  const int j  = blockIdx.x * 256 + threadIdx.x;
  const int i0 = blockIdx.y * 256;

  fk[threadIdx.x] = fmaf(f[pair * N + i0 + threadIdx.x], k2, c2);
  __syncthreads();

  float m = -__builtin_inff(), s = 0.f;
  const float* __restrict__ col = Cp + (size_t)i0 * N + j;
#pragma unroll 4
  for (int ii = 0; ii < 256; ++ii) {
    float v = fmaf(-k2, col[(size_t)ii * N], fk[ii]);
    lse_push(v, m, s);
  }
  const int o = (pair * SPLITS + blockIdx.y) * N + j;
  pm[o] = m; ps[o] = s;
}

// ---------------------------------------------------------------------------
// f-update partial: for row i, reduce columns j in [j0, j0+256).
// Column access transposed through LDS (pitch 33 -> bank-conflict-free on
// 64 banks with wave32). Global loads stay coalesced along j.
// ---------------------------------------------------------------------------
__global__ __launch_bounds__(256) void f_partial_k(
    const float* __restrict__ C, const float* __restrict__ g,
    float* __restrict__ pm, float* __restrict__ ps, float k2, float c2) {
  __shared__ float gk[256];
  __shared__ float tile[256 * 33];                 // 33.8 KB, padded pitch
  const int pair = blockIdx.z;
  const float* __restrict__ Cp = C + (size_t)pair * N * N;
  const int i0  = blockIdx.x * 256;                // rows owned by block
  const int j0  = blockIdx.y * 256;                // column chunk (split)
  const int tid = threadIdx.x;
  const int cl  = tid & 31, rw = tid >> 5;

  gk[tid] = fmaf(g[pair * N + j0 + tid], k2, c2);

  float m = -__builtin_inff(), s = 0.f;
  for (int jt = 0; jt < 256; jt += 32) {
    __syncthreads();                               // also covers gk on jt==0
#pragma unroll
    for (int q = 0; q < 32; ++q) {
      int r = rw + q * 8;                          // coalesced along cl
      tile[r * 33 + cl] = Cp[(size_t)(i0 + r) * N + j0 + jt + cl];
    }
    __syncthreads();
#pragma unroll
    for (int c = 0; c < 32; ++c) {
      float v = fmaf(-k2, tile[tid * 33 + c], gk[jt + c]);
      lse_push(v, m, s);
    }
  }
  const int o = (pair * SPLITS + blockIdx.y) * N + (i0 + tid);
  pm[o] = m; ps[o] = s;
}

// ---------------------------------------------------------------------------
// Merge SPLITS partial (m, s) states per output; write dual potential.
// out[j] = -eps * LSE_e = (m + log2(s)) * (-eps*ln2)
// ---------------------------------------------------------------------------
__global__ __launch_bounds__(256) void combine_k(
    const float* __restrict__ pm, const float* __restrict__ ps,
    float* __restrict__ out, float neg_eps_ln2) {
  const int pair = blockIdx.y;
  const int j = blockIdx.x * 256 + threadIdx.x;
  const int base = pair * SPLITS * N + j;
  float m = pm[base], s = ps[base];
#pragma unroll
  for (int sp = 1; sp < SPLITS; ++sp) {
    float m2 = pm[base + sp * N], s2 = ps[base + sp * N];
    float nm = fmaxf(m, m2);
    s = fmaf(s, exp2f(m - nm), s2 * exp2f(m2 - nm));
    m = nm;
  }
  out[pair * N + j] = (m + log2f(s)) * neg_eps_ln2;
}

// ---------------------------------------------------------------------------
// Final: sum_p (mean f_p + mean g_p) * EPS  ==  (sum all f + g)/N * EPS
// ---------------------------------------------------------------------------
__global__ __launch_bounds__(256) void finalize_k(const float* __restrict__ f,
                                                  const float* __restrict__ g,
                                                  float* __restrict__ out) {
  __shared__ float red[256];
  float acc = 0.f;
  for (int i = threadIdx.x; i < PAIRS * N; i += 256) acc += f[i] + g[i];
  red[threadIdx.x] = acc;
  __syncthreads();
  for (int st = 128; st > 0; st >>= 1) {
    if (threadIdx.x < st) red[threadIdx.x] += red[threadIdx.x + st];
    __syncthreads();
  }
  if (threadIdx.x == 0) out[0] = red[0] * (0.05f / (float)N);
}

// ---------------------------------------------------------------------------
// Host driver: eps schedule computed on host (data-independent), all kernels
// on `stream` (graph-capture safe: no malloc/memcpy/sync).
// Workspace: C (5*N*N f32 = 80 MB) | f | g | pm | ps  ~= 84.7 MB total.
// ---------------------------------------------------------------------------
extern "C" void kernel_launch(void* const* d_in, const int* in_sizes, int n_in,
                              void* d_out, int out_size, void* d_ws, size_t ws_size,
                              hipStream_t stream) {
  (void)in_sizes; (void)n_in; (void)out_size; (void)ws_size;
  const float* x = (const float*)d_in[0];
  float* out = (float*)d_out;

  float* C  = (float*)d_ws;
  float* f  = C  + (size_t)PAIRS * N * N;
  float* g  = f  + PAIRS * N;
  float* pm = g  + PAIRS * N;
  float* ps = pm + PAIRS * SPLITS * N;

  init_fg<<<dim3((PAIRS * N + 255) / 256), 256, 0, stream>>>(f, g);
  build_cost<<<dim3((N / 16) * (N / 16) / 8, PAIRS), 256, 0, stream>>>(x, C);

  const double LOG2E = 1.4426950408889634;
  const double LN2   = 0.6931471805599453;
  for (int t = 0; t < NITERS; ++t) {
    double eps = 100.0 * pow(0.95, 2.0 * t);       // D2_INIT * SCALING^(2t)
    if (eps < 0.05) eps = 0.05;                    // max(EPS, ...)
    float k2 = (float)(LOG2E / eps);               // 1/(eps*ln2)
    float c2 = (float)(-log2((double)N));          // log_a * log2(e) = -log2(n)
    float neg_eps_ln2 = (float)(-eps * LN2);

    // g_j = -eps * LSE_i[(f_i - C_ij)/eps + log a]   (uses current f)
    g_partial_k<<<dim3(N / 256, SPLITS, PAIRS), 256, 0, stream>>>(C, f, pm, ps, k2, c2);
    combine_k<<<dim3(N / 256, PAIRS), 256, 0, stream>>>(pm, ps, g, neg_eps_ln2);
    // f_i = -eps * LSE_j[(g_j - C_ij)/eps + log b]   (uses updated g)
    f_partial_k<<<dim3(N / 256, SPLITS, PAIRS), 256, 0, stream>>>(C, g, pm, ps, k2, c2);
    combine_k<<<dim3(N / 256, PAIRS), 256, 0, stream>>>(pm, ps, f, neg_eps_ln2);
  }

  finalize_k<<<1, 256, 0, stream>>>(f, g, out);
}